// FairPHM_36120674959487
// MI455X (gfx1250) — compile-verified
//
#include <hip/hip_runtime.h>

typedef __attribute__((ext_vector_type(2))) float v2f;
typedef __attribute__((ext_vector_type(8))) float v8f;

#define N_NODES 100000
#define N_EDGES 1600000
#define HID     64
// N_NODES == 6250 * 16 exactly -> no row-boundary checks in the WMMA GEMM.

// ---------------------------------------------------------------------------
// Kernel 1: per-edge variant weight  ew = sigmoid(emb_cat[e] . Wv + bv)
// One wave (32 lanes) per edge; 128 floats = 32 lanes x float4. Also writes
// (1 - ew) into the tail of d_out.
// ---------------------------------------------------------------------------
__global__ void edge_weight_kernel(const float* __restrict__ emb_cat,
                                   const float* __restrict__ Wv,
                                   const float* __restrict__ bv,
                                   float* __restrict__ ew,
                                   float* __restrict__ out_one_minus_ew) {
    int lane = threadIdx.x & 31;
    long long e = (long long)blockIdx.x * (blockDim.x >> 5) + (threadIdx.x >> 5);
    if (e >= N_EDGES) return;
    const float4 a = ((const float4*)(emb_cat + e * 128))[lane];
    const float4 w = ((const float4*)Wv)[lane];
    float dot = a.x * w.x + a.y * w.y + a.z * w.z + a.w * w.w;
    #pragma unroll
    for (int off = 16; off > 0; off >>= 1) dot += __shfl_down(dot, off, 32);
    if (lane == 0) {
        float s = 1.0f / (1.0f + __expf(-(dot + bv[0])));
        ew[e] = s;
        out_one_minus_ew[e] = 1.0f - s;
    }
}

// ---------------------------------------------------------------------------
// Degree / normalization kernels. deg starts at 1.0 (self-loop weight).
// ---------------------------------------------------------------------------
__global__ void init_deg_kernel(float* __restrict__ deg) {
    int i = blockIdx.x * blockDim.x + threadIdx.x;
    if (i < N_NODES) deg[i] = 1.0f;
}

__global__ void scatter_deg_kernel(const int* __restrict__ dst,
                                   const float* __restrict__ ew,
                                   float* __restrict__ deg) {
    int e = blockIdx.x * blockDim.x + threadIdx.x;
    if (e < N_EDGES) atomicAdd(&deg[dst[e]], ew[e]);
}

__global__ void rsqrt_deg_kernel(float* __restrict__ deg_dis) {
    int i = blockIdx.x * blockDim.x + threadIdx.x;
    if (i < N_NODES) {
        float d = deg_dis[i];
        deg_dis[i] = (d > 0.0f) ? rsqrtf(d) : 0.0f;  // deg >= 1 always, kept for parity
    }
}

// ---------------------------------------------------------------------------
// Kernel: out[M,64] = (RELU? relu(x):x)[M,64] @ W[64,64]^T  via
// V_WMMA_F32_16X16X4_F32 (fp32 A/B/C: exact vs reference).
// Block = 128 threads = 4 waves. Block b owns rows [16b,16b+16); wave w owns
// cols [16w,16w+16). K=64 -> 16 chained WMMAs, accumulator in registers.
//
// Fragment layouts per CDNA5 ISA §7.12.2:
//   A 16x4 f32: VGPR0/1 = K{k0,k0+1} on lanes 0-15 (M=lane),
//               K{k0+2,k0+3} on lanes 16-31.
//   B  4x16:    mirrored; B[k][n] = W[n*64+k] (x@W^T needs no transpose).
//   C/D v8f:    VGPR j -> row = base + j + 8*(lane>=16), col = base + (lane&15).
// ---------------------------------------------------------------------------
template <bool RELU>
__global__ void gemm_xWT_wmma_kernel(const float* __restrict__ x,
                                     const float* __restrict__ W,
                                     float* __restrict__ out) {
    const int wave = threadIdx.x >> 5;
    const int lane = threadIdx.x & 31;
    const int hi   = lane >> 4;    // 0: K pair {0,1}; 1: K pair {2,3}
    const int lo   = lane & 15;
    const int row_base = blockIdx.x * 16;
    const int col_base = wave * 16;

    v8f acc = {};
    #pragma unroll
    for (int kk = 0; kk < 16; ++kk) {
        const int k0 = kk * 4 + 2 * hi;
        const int m = row_base + lo;           // A: M = lo
        float a0 = x[m * HID + k0];
        float a1 = x[m * HID + k0 + 1];
        if (RELU) { a0 = fmaxf(a0, 0.0f); a1 = fmaxf(a1, 0.0f); }
        v2f a; a.x = a0; a.y = a1;

        const int n = col_base + lo;           // B: N = lo
        v2f b;
        b.x = W[n * HID + k0];
        b.y = W[n * HID + k0 + 1];

        acc = __builtin_amdgcn_wmma_f32_16x16x4_f32(
            /*neg_a=*/false, a, /*neg_b=*/false, b,
            /*c_mod=*/(short)0, acc, /*reuse_a=*/false, /*reuse_b=*/false);
    }
    #pragma unroll
    for (int j = 0; j < 8; ++j)
        out[(row_base + j + 8 * hi) * HID + col_base + lo] = acc[j];
}

// ---------------------------------------------------------------------------
// Aggregation: agg[n,:] = dis[n]^2 * xW[n,:] + b  (self-loop, weight 1) then
// edge scatter: agg[dst,:] += dis[src]*ew*dis[dst] * xW[src,:]
// ---------------------------------------------------------------------------
__global__ void agg_init_kernel(const float* __restrict__ xW,
                                const float* __restrict__ dis,
                                const float* __restrict__ b,
                                float* __restrict__ agg) {
    long long i = (long long)blockIdx.x * blockDim.x + threadIdx.x;
    if (i < (long long)N_NODES * HID) {
        int n = (int)(i >> 6), c = (int)(i & 63);
        float d = dis[n];
        agg[i] = d * d * xW[i] + b[c];
    }
}

__global__ void agg_edges_kernel(const int* __restrict__ src,
                                 const int* __restrict__ dst,
                                 const float* __restrict__ ew,
                                 const float* __restrict__ dis,
                                 const float* __restrict__ xW,
                                 float* __restrict__ agg) {
    long long i = (long long)blockIdx.x * blockDim.x + threadIdx.x;
    if (i < (long long)N_EDGES * HID) {
        long long e = i >> 6;
        int c = (int)(i & 63);
        int s = src[e], d = dst[e];
        float coef = dis[s] * ew[e] * dis[d];
        atomicAdd(&agg[(long long)d * HID + c], coef * xW[(long long)s * HID + c]);
    }
}

// ---------------------------------------------------------------------------
// Classifier: logits = [h | labels] @ Wc^T + bc ; 2-class softmax.
// One wave per node: lanes cover the 64 hidden channels, shuffle-reduce.
// ---------------------------------------------------------------------------
__global__ void classifier_kernel(const float* __restrict__ h,
                                  const float* __restrict__ labels,
                                  const float* __restrict__ Wc,  // [2, 65]
                                  const float* __restrict__ bc,  // [2]
                                  float* __restrict__ out) {     // [N, 2]
    int lane = threadIdx.x & 31;
    int n = blockIdx.x * (blockDim.x >> 5) + (threadIdx.x >> 5);
    if (n >= N_NODES) return;
    float h0 = h[n * HID + lane];
    float h1 = h[n * HID + 32 + lane];
    float l0 = h0 * Wc[lane]      + h1 * Wc[32 + lane];
    float l1 = h0 * Wc[65 + lane] + h1 * Wc[65 + 32 + lane];
    #pragma unroll
    for (int off = 16; off > 0; off >>= 1) {
        l0 += __shfl_down(l0, off, 32);
        l1 += __shfl_down(l1, off, 32);
    }
    if (lane == 0) {
        float lab = labels[n];
        l0 += lab * Wc[64]      + bc[0];
        l1 += lab * Wc[65 + 64] + bc[1];
        float m  = fmaxf(l0, l1);
        float e0 = __expf(l0 - m), e1 = __expf(l1 - m);
        float inv = 1.0f / (e0 + e1);
        out[n * 2]     = e0 * inv;
        out[n * 2 + 1] = e1 * inv;
    }
}

// ---------------------------------------------------------------------------
// Launch. Inputs (setup_inputs order):
//  0 emb_cat[E,128] 1 features[N,64] 2 labels[N] 3 Wv[1,128] 4 bv[1]
//  5 W1[64,64] 6 b1[64] 7 W2[64,64] 8 b2[64] 9 Wc[2,65] 10 bc[2]
//  11 edge_index[2,E] (int32)
// d_out = sens[N,2] ++ (1-ew)[E]  (float32)
// ---------------------------------------------------------------------------
extern "C" void kernel_launch(void* const* d_in, const int* in_sizes, int n_in,
                              void* d_out, int out_size, void* d_ws, size_t ws_size,
                              hipStream_t stream) {
    const float* emb_cat  = (const float*)d_in[0];
    const float* features = (const float*)d_in[1];
    const float* labels   = (const float*)d_in[2];
    const float* Wv       = (const float*)d_in[3];
    const float* bv       = (const float*)d_in[4];
    const float* W1       = (const float*)d_in[5];
    const float* b1       = (const float*)d_in[6];
    const float* W2       = (const float*)d_in[7];
    const float* b2       = (const float*)d_in[8];
    const float* Wc       = (const float*)d_in[9];
    const float* bc       = (const float*)d_in[10];
    const int*   src      = (const int*)d_in[11];
    const int*   dst      = src + N_EDGES;

    float* out_sens = (float*)d_out;                       // N*2
    float* out_1mew = out_sens + 2 * N_NODES;              // E

    // Workspace carve-up (floats): ew | deg/dis | buf0 | buf1
    float* ew   = (float*)d_ws;
    float* dis  = ew  + N_EDGES;
    float* buf0 = dis + N_NODES;                           // x@W^T results
    float* buf1 = buf0 + (size_t)N_NODES * HID;            // aggregated h

    // 1) edge weights (+ output tail), 8 edges per 256-thread block
    edge_weight_kernel<<<N_EDGES / 8, 256, 0, stream>>>(emb_cat, Wv, bv, ew, out_1mew);

    // 2) symmetric normalization coefficients
    init_deg_kernel<<<(N_NODES + 255) / 256, 256, 0, stream>>>(dis);
    scatter_deg_kernel<<<N_EDGES / 256, 256, 0, stream>>>(dst, ew, dis);
    rsqrt_deg_kernel<<<(N_NODES + 255) / 256, 256, 0, stream>>>(dis);

    const int gemm_blocks = N_NODES / 16;                  // 6250, exact
    const long long ncell = (long long)N_NODES * HID;
    const long long ecell = (long long)N_EDGES * HID;
    const int agg_init_blocks  = (int)((ncell + 255) / 256);
    const int agg_edge_blocks  = (int)((ecell + 255) / 256);

    // 3) layer 1: WMMA transform -> self-loop+bias init -> edge scatter
    gemm_xWT_wmma_kernel<false><<<gemm_blocks, 128, 0, stream>>>(features, W1, buf0);
    agg_init_kernel<<<agg_init_blocks, 256, 0, stream>>>(buf0, dis, b1, buf1);
    agg_edges_kernel<<<agg_edge_blocks, 256, 0, stream>>>(src, dst, ew, dis, buf0, buf1);

    // 4) layer 2: ReLU fused into WMMA A-load; ping-pong buffers
    gemm_xWT_wmma_kernel<true><<<gemm_blocks, 128, 0, stream>>>(buf1, W2, buf0);
    agg_init_kernel<<<agg_init_blocks, 256, 0, stream>>>(buf0, dis, b2, buf1);
    agg_edges_kernel<<<agg_edge_blocks, 256, 0, stream>>>(src, dst, ew, dis, buf0, buf1);

    // 5) classifier + softmax (8 nodes per 256-thread block)
    classifier_kernel<<<(N_NODES + 7) / 8, 256, 0, stream>>>(buf1, labels, Wc, bc, out_sens);
}